// ReinforceAgentQuantum_32822140076702
// MI455X (gfx1250) — compile-verified
//
#include <hip/hip_runtime.h>
#include <math.h>

#define NQ 14
#define NS 16384         // 2^14 amplitudes
#define NPAIR 91         // C(14,2)
#define NL 2
#define NA 14
#define NTHREADS 256
#define FPT (NS / NTHREADS)   // 64 amplitudes per thread

typedef __attribute__((ext_vector_type(2))) float v2f;
typedef __attribute__((ext_vector_type(8))) float v8f;

// basis index from (outer index, 4-bit group value, group bit-shift)
__device__ __forceinline__ int faddr(int o, int j, int gs) {
  int lowmask = (1 << gs) - 1;
  return ((o & ~lowmask) << 4) | (j << gs) | (o & lowmask);
}

// Element (i,j) of the fused 4-qubit RX matrix  U = prod_s (c I + (-i s) X).
// Slot s <-> state bit (gs+s) <-> qubit 13-(gs+s). Inactive slots are identity.
__device__ __forceinline__ void u_elem(int i, int j, int gs, int act,
                                       const float* cs, const float* ss,
                                       float& ur, float& uim) {
  float mag = 1.0f; int nd = 0; bool ok = true;
#pragma unroll
  for (int s = 0; s < 4; ++s) {
    int bi = (i >> s) & 1, bj = (j >> s) & 1;
    if ((act >> s) & 1) {
      int q = 13 - (gs + s);
      if (bi == bj) mag *= cs[q]; else { mag *= ss[q]; ++nd; }
    } else if (bi != bj) ok = false;
  }
  if (!ok) mag = 0.0f;
  // multiply by (-i)^nd
  ur = 0.0f; uim = 0.0f;
  if ((nd & 1) == 0) ur  = (nd & 2) ? -mag : mag;
  else               uim = (nd & 2) ?  mag : -mag;
}

#define WMMA4(a, b, c) \
  __builtin_amdgcn_wmma_f32_16x16x4_f32(false, (a), false, (b), (short)0, (c), false, false)

__global__ __launch_bounds__(NTHREADS)
void qsim_kernel(const float* __restrict__ x,       // [B,42]
                 const float* __restrict__ iscale,  // [2]
                 const float* __restrict__ w,       // [2]
                 const float* __restrict__ oscale,  // [14]
                 float* __restrict__ out) {         // [B,14]
  __shared__ float2 st[NS];               // interleaved (real, imag) -> b64 LDS ops
  __shared__ float xl[3 * NA];
  __shared__ float dpair[NPAIR];
  __shared__ float dsc[NPAIR];
  __shared__ int   iiL[NPAIR], jjL[NPAIR];
  __shared__ float Phh[128], Pll[128], Mx[7 * 128];
  __shared__ float csL[NL][NQ], ssL[NL][NQ];
  __shared__ float ev[NA];

  const int tid = threadIdx.x;
  const int b   = blockIdx.x;

  // ---------------- setup ----------------
  if (tid < 3 * NA) xl[tid] = x[b * (3 * NA) + tid];
  if (tid < NA)     ev[tid] = 0.0f;
  __syncthreads();
  if (tid < NPAIR) {                       // pair list + distances
    int i = 0, rem = tid;
    while (rem >= 13 - i) { rem -= 13 - i; ++i; }
    int j = i + 1 + rem;
    iiL[tid] = i; jjL[tid] = j;
    dpair[tid] = fabsf(xl[NA + i] - xl[NA + j]);
  }
  if (tid < NL * NQ) {                     // RX half-angle cos/sin per layer
    int l = tid / NQ, q = tid % NQ;
    float ann = (xl[q] > 0.0f) ? 3.14159265358979323846f : 0.0f;
    float ph  = 0.5f * w[l] * ann;
    csL[l][q] = cosf(ph); ssL[l][q] = sinf(ph);
  }
  const float amp0 = 0.0078125f;           // 2^(-14/2)
  for (int k = 0; k < FPT; ++k) { int f = tid + NTHREADS * k; st[f] = make_float2(amp0, 0.0f); }
  __syncthreads();

  for (int layer = 0; layer < NL; ++layer) {
    if (tid < NPAIR) dsc[tid] = iscale[layer] * dpair[tid];
    __syncthreads();

    // ---- ZZ-angle decomposition tables:  angle(h,l)=Phh[h]+Pll[l]+sum_i z_i(h)*Mx[i][l]
    for (int e = tid; e < 128 + 128 + 7 * 128; e += NTHREADS) {
      if (e < 128) {
        int h = e; float a = 0.0f;
        for (int p = 0; p < NPAIR; ++p) {
          int i = iiL[p], j = jjL[p];
          if (j <= 6) {
            float zi = ((h >> (6 - i)) & 1) ? -1.0f : 1.0f;
            float zj = ((h >> (6 - j)) & 1) ? -1.0f : 1.0f;
            a += zi * zj * dsc[p];
          }
        }
        Phh[h] = a;
      } else if (e < 256) {
        int lo = e - 128; float a = 0.0f;
        for (int p = 0; p < NPAIR; ++p) {
          int i = iiL[p], j = jjL[p];
          if (i >= 7) {
            float zi = ((lo >> (13 - i)) & 1) ? -1.0f : 1.0f;
            float zj = ((lo >> (13 - j)) & 1) ? -1.0f : 1.0f;
            a += zi * zj * dsc[p];
          }
        }
        Pll[lo] = a;
      } else {
        int em = e - 256; int iq = em >> 7; int lo = em & 127;
        float a = 0.0f;
        for (int p = 0; p < NPAIR; ++p) {
          int i = iiL[p], j = jjL[p];
          if (i == iq && j >= 7) {
            float zj = ((lo >> (13 - j)) & 1) ? -1.0f : 1.0f;
            a += zj * dsc[p];
          }
        }
        Mx[em] = a;
      }
    }
    __syncthreads();

    // ---- diagonal phase:  state *= exp(-0.5i * angle)
    for (int k = 0; k < FPT; ++k) {
      int f = tid + NTHREADS * k;
      int h = f >> 7, lo = f & 127;
      float a = Phh[h] + Pll[lo];
#pragma unroll
      for (int i = 0; i < 7; ++i) {
        float m = Mx[i * 128 + lo];
        a += ((h >> (6 - i)) & 1) ? -m : m;
      }
      float th = -0.5f * a;
      float sn, cn; __sincosf(th, &sn, &cn);
      float2 v = st[f];
      st[f] = make_float2(v.x * cn - v.y * sn, v.x * sn + v.y * cn);
    }
    __syncthreads();

    // ---- RX layer: 4 fused 4-qubit groups, complex 16x16 GEMM via v_wmma_f32_16x16x4_f32
    const int lane = tid & 31;
    const int wv   = tid >> 5;
    const int mrow = lane & 15;
    const int hi   = lane >> 4;
#pragma unroll 1
    for (int g = 0; g < 4; ++g) {
      const int gs  = (g == 0) ? 0 : (g == 1) ? 4 : (g == 2) ? 8 : 10;
      const int act = (g == 3) ? 0xC : 0xF;   // last group: RX only on qubits 0,1

      // A fragments (U real, U imag, -U imag), K-slices of 4
      v2f aUr[4], aUi[4], aUiN[4];
#pragma unroll
      for (int kk = 0; kk < 4; ++kk) {
#pragma unroll
        for (int v = 0; v < 2; ++v) {
          int K = 4 * kk + 2 * hi + v;     // ISA A layout: lane-half selects K+2
          float ur, uim;
          u_elem(mrow, K, gs, act, &csL[layer][0], &ssL[layer][0], ur, uim);
          aUr[kk][v] = ur; aUi[kk][v] = uim; aUiN[kk][v] = -uim;
        }
      }

#pragma unroll 1
      for (int t = 0; t < 8; ++t) {        // each wave owns 8 column tiles
        const int ob = wv * 128 + t * 16 + mrow;  // outer index for this lane's column
        v2f bR[4], bI[4];
#pragma unroll
        for (int kk = 0; kk < 4; ++kk) {
#pragma unroll
          for (int v = 0; v < 2; ++v) {
            int j = 4 * kk + 2 * hi + v;
            float2 q = st[faddr(ob, j, gs)];   // single ds_load_b64
            bR[kk][v] = q.x; bI[kk][v] = q.y;
          }
        }
        v8f dR = {0, 0, 0, 0, 0, 0, 0, 0};
        v8f dI = {0, 0, 0, 0, 0, 0, 0, 0};
#pragma unroll
        for (int kk = 0; kk < 4; ++kk) {
          dR = WMMA4(aUr[kk],  bR[kk], dR);   // Dr += Ur*Sr
          dR = WMMA4(aUiN[kk], bI[kk], dR);   // Dr -= Ui*Si
          dI = WMMA4(aUr[kk],  bI[kk], dI);   // Di += Ur*Si
          dI = WMMA4(aUi[kk],  bR[kk], dI);   // Di += Ui*Sr
        }
#pragma unroll
        for (int v = 0; v < 8; ++v) {
          int Mr = v + 8 * hi;               // ISA C/D layout
          st[faddr(ob, Mr, gs)] = make_float2(dR[v], dI[v]);  // single ds_store_b64
        }
      }
      __syncthreads();
    }
  }

  // ---------------- <X_q> and output ----------------
  float acc[NA];
#pragma unroll
  for (int q = 0; q < NA; ++q) acc[q] = 0.0f;
  for (int k = 0; k < FPT; ++k) {
    int f = tid + NTHREADS * k;
    float2 a = st[f];
#pragma unroll
    for (int q = 0; q < NA; ++q) {
      float2 p = st[f ^ (1 << (13 - q))];    // single ds_load_b64 per partner
      acc[q] += a.x * p.x + a.y * p.y;
    }
  }
#pragma unroll
  for (int q = 0; q < NA; ++q) atomicAdd(&ev[q], acc[q]);
  __syncthreads();
  if (tid < NA) out[b * NA + tid] = ev[tid] * oscale[tid];
}

extern "C" void kernel_launch(void* const* d_in, const int* in_sizes, int n_in,
                              void* d_out, int out_size, void* d_ws, size_t ws_size,
                              hipStream_t stream) {
  const float* x      = (const float*)d_in[0];
  const float* iscale = (const float*)d_in[1];
  const float* w      = (const float*)d_in[2];
  const float* oscale = (const float*)d_in[3];
  float* out = (float*)d_out;
  const int B = in_sizes[0] / (3 * NA);   // 1024
  qsim_kernel<<<B, NTHREADS, 0, stream>>>(x, iscale, w, oscale, out);
}